// GCN_54786602828281
// MI455X (gfx1250) — compile-verified
//
#include <hip/hip_runtime.h>
#include <hip/hip_bf16.h>

typedef __attribute__((ext_vector_type(2))) float v2f;
typedef __attribute__((ext_vector_type(8))) float v8f;

static __host__ inline int cdiv(int a, int b) { return (a + b - 1) / b; }

// ---------------- small elementwise kernels ----------------

__global__ void k_fill(float* __restrict__ p, float v, int n) {
    int i = blockIdx.x * blockDim.x + threadIdx.x;
    if (i < n) p[i] = v;
}

__global__ void k_deg_accum(const int* __restrict__ dst, float* __restrict__ deg, int E) {
    int e = blockIdx.x * blockDim.x + threadIdx.x;
    if (e < E) atomicAdd(&deg[dst[e]], 1.0f);
}

__global__ void k_rsqrt(float* __restrict__ d, int n) {
    int i = blockIdx.x * blockDim.x + threadIdx.x;
    if (i < n) {
        float v = d[i];
        d[i] = (v > 0.0f) ? rsqrtf(v) : 0.0f;
    }
}

// Bt[ncol][k] = (ncol < Ncols) ? Bw[k][ncol] : 0   -- transpose + zero-pad to ntiles*16 cols
__global__ void k_transpose_pad(const float* __restrict__ Bw, float* __restrict__ Bt,
                                int K, int Ncols, int ncols_pad) {
    int i = blockIdx.x * blockDim.x + threadIdx.x;
    int total = ncols_pad * K;
    if (i >= total) return;
    int ncol = i / K, k = i - ncol * K;
    Bt[i] = (ncol < Ncols) ? Bw[(size_t)k * Ncols + ncol] : 0.0f;
}

// One thread per edge: agg[dst] += t[src] * dinv[src]*dinv[dst]  (20 feats, 5x float4 loads)
__global__ void k_edge_scatter(const int* __restrict__ src, const int* __restrict__ dst,
                               const float* __restrict__ t, const float* __restrict__ dinv,
                               float* __restrict__ agg, int E) {
    int e = blockIdx.x * blockDim.x + threadIdx.x;
    if (e >= E) return;
    int s = src[e], d = dst[e];
    float nr = dinv[s] * dinv[d];
    const float4* ts = (const float4*)(t + (size_t)s * 20);
    float* ad = agg + (size_t)d * 20;
#pragma unroll
    for (int q = 0; q < 5; ++q) {
        float4 v = ts[q];
        atomicAdd(ad + 4 * q + 0, v.x * nr);
        atomicAdd(ad + 4 * q + 1, v.y * nr);
        atomicAdd(ad + 4 * q + 2, v.z * nr);
        atomicAdd(ad + 4 * q + 3, v.w * nr);
    }
}

// h = relu(agg + t * dinv^2 (self loop) + bias[f])   -- written in place into agg
__global__ void k_finish_layer(const float* __restrict__ t, const float* __restrict__ dinv,
                               const float* __restrict__ bias, float* __restrict__ agg, int n_nodes) {
    int i = blockIdx.x * blockDim.x + threadIdx.x;
    int total = n_nodes * 20;
    if (i >= total) return;
    int node = i / 20, f = i - node * 20;
    float di = dinv[node];
    float v = agg[i] + t[i] * di * di + bias[f];
    agg[i] = fmaxf(v, 0.0f);
}

// ---------------- fp32 WMMA GEMM ----------------
// C[M,Ncols] = A[M,K] @ B[K,Ncols] (+bias)(+relu), with B pre-transposed/padded:
// Bt is [ntiles*16, K] row-major. K must be a multiple of 4 (20 / 1640 / 300 all are).
// One wave per (16-row M-tile, 16-col N-tile); waves in a block share the N-tile (B in WGP$).
// Fragment layouts per CDNA5 ISA 7.12.2 (32-bit, wave32):
//   A 16x4 : lane&15 = M row; VGPR0 = K (lane<16 ? kb+0 : kb+2), VGPR1 = +1  -> one b64/lane
//   B 4x16 : lane&15 = N col; same K-per-lane-half striping                  -> one b64/lane
//   C/D    : lane&15 = N col; VGPR r = M row r (lanes 0-15) / r+8 (lanes 16-31)
__global__ void k_wmma_gemm_f32(const float* __restrict__ A, const float* __restrict__ Bt,
                                const float* __restrict__ bias, float* __restrict__ C,
                                int M, int K, int Ncols, int mtiles, int ntiles, int do_relu) {
    int wave = (blockIdx.x * blockDim.x + threadIdx.x) >> 5;
    int lane = threadIdx.x & 31;
    int mt = wave % mtiles;                  // consecutive waves: same nt, different mt
    int nt = wave / mtiles;
    if (nt >= ntiles) return;                // wave-uniform: EXEC all-ones at every WMMA

    int lrow = lane & 15;                    // M row (A) / N col (B, C/D)
    int kh = (lane >> 4) << 1;               // 0 or 2 per lane half
    int ncol = nt * 16 + lrow;
    const float* arow = A + (size_t)(mt * 16 + lrow) * K + kh;   // 8B aligned (K even)
    const float* brow = Bt + (size_t)ncol * K + kh;              // 8B aligned

    v8f acc = {};
    int kb = 0;
    for (; kb + 16 <= K; kb += 16) {
#pragma unroll
        for (int j = 0; j < 4; ++j) {
            v2f a = *(const v2f*)(arow + kb + 4 * j);
            v2f b = *(const v2f*)(brow + kb + 4 * j);
            acc = __builtin_amdgcn_wmma_f32_16x16x4_f32(false, a, false, b,
                                                        (short)0, acc, false, false);
        }
    }
    for (; kb < K; kb += 4) {
        v2f a = *(const v2f*)(arow + kb);
        v2f b = *(const v2f*)(brow + kb);
        acc = __builtin_amdgcn_wmma_f32_16x16x4_f32(false, a, false, b,
                                                    (short)0, acc, false, false);
    }

    if (ncol >= Ncols) return;               // only tail N-tile partially stores
    float bv = bias ? bias[ncol] : 0.0f;
    int rbase = mt * 16 + ((lane >> 4) << 3);
#pragma unroll
    for (int r = 0; r < 8; ++r) {
        float v = acc[r] + bv;
        if (do_relu) v = fmaxf(v, 0.0f);
        C[(size_t)(rbase + r) * Ncols + ncol] = v;
    }
}

// ---------------- launch ----------------

extern "C" void kernel_launch(void* const* d_in, const int* in_sizes, int n_in,
                              void* d_out, int out_size, void* d_ws, size_t ws_size,
                              hipStream_t stream) {
    const float* x    = (const float*)d_in[0];
    const int*   eidx = (const int*)d_in[1];   // [2, E] flat
    /* d_in[2] = batch_index: unused by the reference computation */
    const float* W1   = (const float*)d_in[3];
    const float* b1   = (const float*)d_in[4];
    const float* W2   = (const float*)d_in[5];
    const float* b2   = (const float*)d_in[6];
    const float* fc1W = (const float*)d_in[7];
    const float* fc1b = (const float*)d_in[8];
    const float* outW = (const float*)d_in[9];
    const float* outb = (const float*)d_in[10];

    const int N  = in_sizes[2];          // total nodes
    const int E  = in_sizes[1] / 2;      // directed edges
    const int Bg = N / 82;               // graphs
    const int EMB = 20, LAT = 82 * 20, HID = 300, NC = 22;

    // workspace layout (256B aligned slabs)
    char* ws = (char*)d_ws;
    size_t off = 0;
    auto take = [&](size_t bytes) -> void* {
        off = (off + 255) & ~(size_t)255;
        void* p = ws + off;
        off += bytes;
        return p;
    };
    float* dinv = (float*)take((size_t)N * 4);            // deg -> dinv in place
    float* t    = (float*)take((size_t)N * EMB * 4);      // x@W / h@W
    float* agg  = (float*)take((size_t)N * EMB * 4);      // scatter target / h
    float* fco  = (float*)take((size_t)Bg * HID * 4);     // fc1 activations
    float* Bt   = (float*)take((size_t)304 * LAT * 4);    // transposed/padded weights (reused)

    const int* src = eidx;
    const int* dst = eidx + E;

    const int TB = 256;
    int nfeat = N * EMB;

    auto run_gemm = [&](const float* A, const float* Bw, const float* bias, float* C,
                        int M, int K, int Ncols, int relu) {
        int ntiles = (Ncols + 15) / 16;
        int ncols_pad = ntiles * 16;
        int mtiles = M / 16;                               // M is a multiple of 16 here
        k_transpose_pad<<<cdiv(ncols_pad * K, TB), TB, 0, stream>>>(Bw, Bt, K, Ncols, ncols_pad);
        int waves = mtiles * ntiles;
        k_wmma_gemm_f32<<<cdiv(waves * 32, TB), TB, 0, stream>>>(
            A, Bt, bias, C, M, K, Ncols, mtiles, ntiles, relu);
    };

    // degrees (self loop contributes 1 to every node) -> dinv
    k_fill<<<cdiv(N, TB), TB, 0, stream>>>(dinv, 1.0f, N);
    k_deg_accum<<<cdiv(E, TB), TB, 0, stream>>>(dst, dinv, E);
    k_rsqrt<<<cdiv(N, TB), TB, 0, stream>>>(dinv, N);

    // ---- GCN layer 1 ----
    k_fill<<<cdiv(nfeat, TB), TB, 0, stream>>>(agg, 0.0f, nfeat);
    run_gemm(x, W1, nullptr, t, N, 20, EMB, 0);                     // t = x @ W1
    k_edge_scatter<<<cdiv(E, TB), TB, 0, stream>>>(src, dst, t, dinv, agg, E);
    k_finish_layer<<<cdiv(nfeat, TB), TB, 0, stream>>>(t, dinv, b1, agg, N);   // h1 in agg

    // ---- GCN layer 2 ----
    run_gemm(agg, W2, nullptr, t, N, EMB, EMB, 0);                  // t = h1 @ W2
    k_fill<<<cdiv(nfeat, TB), TB, 0, stream>>>(agg, 0.0f, nfeat);   // ordered after gemm
    k_edge_scatter<<<cdiv(E, TB), TB, 0, stream>>>(src, dst, t, dinv, agg, E);
    k_finish_layer<<<cdiv(nfeat, TB), TB, 0, stream>>>(t, dinv, b2, agg, N);   // h2 in agg

    // ---- fc1: [Bg, 1640] @ [1640, 300] + bias, relu ----
    run_gemm(agg /* latent view */, fc1W, fc1b, fco, Bg, LAT, HID, 1);

    // ---- out: [Bg, 300] @ [300, 22] + bias ----
    run_gemm(fco, outW, outb, (float*)d_out, Bg, HID, NC, 0);
}